// FftModule_10849087390401
// MI455X (gfx1250) — compile-verified
//
#include <hip/hip_runtime.h>
#include <math.h>

// ---------------------------------------------------------------------------
// FFT of N = 2^26 real f32 -> complex, output planar [2, N] (re plane, im plane)
// 3-pass Cooley-Tukey: N = N1*N2*N3 = 256*256*1024, memory-bound design:
//   pass1: 256-pt FFTs over stride M=2^18 (+ W_N twiddle)      in  -> ws
//   pass2: 256-pt FFTs over stride 1024  (+ W_M twiddle)       ws  -> ws (in place)
//   pass3: 1024-pt FFTs contiguous (= 4x stride-4 256-FFT + radix-4), write
//          transposed planar output                            ws  -> out
// All 256-pt FFTs run per-wave on V_WMMA_F32_16X16X4_F32; tile fills for
// passes 2/3 use GLOBAL_LOAD_ASYNC_TO_LDS (ASYNCcnt) — CDNA5-only data path.
// ---------------------------------------------------------------------------

typedef __attribute__((ext_vector_type(2))) float v2f;
typedef __attribute__((ext_vector_type(8))) float v8f;

#define FFT_N   (1u << 26)
#define N1      256u
#define N2      256u
#define N3      1024u
#define MDIM    (N2 * N3)          /* 2^18 */
#define CB      32u                /* columns per workgroup, passes 1-2 */
#define LLEN    256u
#define TPB     256u               /* 8 waves */
#define P3_LINES 8u                /* lines per workgroup, pass 3 */

__device__ __forceinline__ v8f wmma_f32(v2f a, v2f b, v8f c) {
  // D = A(16x4,f32) * B(4x16,f32) + C(16x16,f32); 8-arg form, no A/B neg for f32.
  return __builtin_amdgcn_wmma_f32_16x16x4_f32(false, a, false, b, (short)0, c,
                                               false, false);
}

// Low 32 bits of a generic shared-memory pointer = wave-relative LDS byte
// address (ISA 10.2: LDS aperture addresses truncate to addr[31:0]).
__device__ __forceinline__ unsigned lds_addr32(const void* p) {
  return (unsigned)(unsigned long long)p;
}

// ---------------------------------------------------------------------------
// Per-lane constants for the WMMA 256-FFT (depend only on lane id):
//   fR/fI/fIn[t]: F16 entries W16^{lo*(4t+2hi+v)} (A-operand stage A; by the
//                 symmetry F[m][k]=F[k][m] also the B-operand of stage C)
//   twc/tws[r]  : intra-256 twiddle W256^{p*lo}, p = r + 8*hi
// ---------------------------------------------------------------------------
struct WaveConsts {
  v2f fR[4], fI[4], fIn[4];
  float twc[8], tws[8];
};

__device__ __forceinline__ WaveConsts make_wave_consts() {
  WaveConsts wc;
  const unsigned lane = threadIdx.x & 31u;
  const unsigned lo = lane & 15u;
  const unsigned hi = lane >> 4;
  const float W16a  = -6.283185307179586f / 16.0f;
  const float W256a = -6.283185307179586f / 256.0f;
#pragma unroll
  for (unsigned t = 0; t < 4; ++t) {
    const unsigned k0 = 4u * t + 2u * hi;
    float s0, c0, s1, c1;
    __sincosf(W16a * (float)((lo * k0) & 15u), &s0, &c0);
    __sincosf(W16a * (float)((lo * (k0 + 1u)) & 15u), &s1, &c1);
    wc.fR[t][0] = c0;  wc.fR[t][1] = c1;
    wc.fI[t][0] = s0;  wc.fI[t][1] = s1;
    wc.fIn[t][0] = -s0; wc.fIn[t][1] = -s1;
  }
#pragma unroll
  for (unsigned r = 0; r < 8; ++r) {
    const unsigned p = r + 8u * hi;
    __sincosf(W256a * (float)((p * lo) & 255u), &wc.tws[r], &wc.twc[r]);
  }
  return wc;
}

// ---------------------------------------------------------------------------
// Per-wave 256-point complex FFT over LDS, element a at lds[idx(a)].
// Four-step: 256 = 16x16, a = i*16 + j, result index k = p + 16q.
//   Stage A: Y = F16 * M          (16 WMMAs)
//   twiddle: Y' = Y .* W256^{p*j}, transpose-store to LDS
//   Stage C: Z = Y' * F16         (16 WMMAs), store Z[p][q] at a' = p + 16q
// Register layouts per ISA 7.12.2 (wave32):
//   A 16x4 : lane L holds row m = L&15, reg v holds K = 2*(L>>4) + v (per chunk)
//   B 4x16 : lane L holds col n = L&15, reg v holds row K = 2*(L>>4) + v
//   C 16x16: lane L holds col n = L&15, reg r holds row m = r + 8*(L>>4)
// ---------------------------------------------------------------------------
template <typename IndexFn>
__device__ __forceinline__ void fft256_wmma(float2* lds, const WaveConsts& wc,
                                            IndexFn idx) {
  const unsigned lane = threadIdx.x & 31u;
  const unsigned lo = lane & 15u;
  const unsigned hi = lane >> 4;

  v8f yR = {0, 0, 0, 0, 0, 0, 0, 0};
  v8f yI = {0, 0, 0, 0, 0, 0, 0, 0};

#pragma unroll
  for (unsigned t = 0; t < 4; ++t) {
    const unsigned k0 = 4u * t + 2u * hi;
    // B = M: M[i][j] = data[i*16+j], rows i = k0,k0+1, col j = lo
    float2 m0 = lds[idx((k0)*16u + lo)];
    float2 m1 = lds[idx((k0 + 1u) * 16u + lo)];
    v2f bR = {m0.x, m1.x};
    v2f bI = {m0.y, m1.y};
    yR = wmma_f32(wc.fR[t], bR, yR);   // Fr*Mr
    yR = wmma_f32(wc.fIn[t], bI, yR);  // - Fi*Mi
    yI = wmma_f32(wc.fR[t], bI, yI);   // Fr*Mi
    yI = wmma_f32(wc.fI[t], bR, yI);   // + Fi*Mr
  }

  // Twiddle W256^{p*j} (p = r + 8*hi, j = lo) and store Y' at element p*16+j.
#pragma unroll
  for (unsigned r = 0; r < 8; ++r) {
    const unsigned p = r + 8u * hi;
    float2 o;
    o.x = yR[r] * wc.twc[r] - yI[r] * wc.tws[r];
    o.y = yR[r] * wc.tws[r] + yI[r] * wc.twc[r];
    lds[idx(p * 16u + lo)] = o;
  }
  // RAW through LDS within the wave (also a compiler reordering barrier).
  asm volatile("s_wait_dscnt 0" ::: "memory");

  v8f zR = {0, 0, 0, 0, 0, 0, 0, 0};
  v8f zI = {0, 0, 0, 0, 0, 0, 0, 0};

#pragma unroll
  for (unsigned t = 0; t < 4; ++t) {
    const unsigned k0 = 4u * t + 2u * hi;
    // A = Y': row m = lo, K = k0,k0+1 -> element lo*16 + K (contiguous pair)
    float2 y0 = lds[idx(lo * 16u + k0)];
    float2 y1 = lds[idx(lo * 16u + k0 + 1u)];
    v2f aR  = {y0.x, y1.x};
    v2f aI  = {y0.y, y1.y};
    v2f aIn = {-y0.y, -y1.y};
    // B = F16 (same per-lane values as stage-A A operand by symmetry).
    zR = wmma_f32(aR, wc.fR[t], zR);
    zR = wmma_f32(aIn, wc.fI[t], zR);
    zI = wmma_f32(aR, wc.fI[t], zI);
    zI = wmma_f32(aI, wc.fR[t], zI);
  }

  // Store Z[p][q] to natural-order result index a' = p + 16*q  (q = lo).
#pragma unroll
  for (unsigned r = 0; r < 8; ++r) {
    const unsigned p = r + 8u * hi;
    lds[idx(p + 16u * lo)] = make_float2(zR[r], zI[r]);
  }
  asm volatile("s_wait_dscnt 0" ::: "memory");
}

// LDS bank-conflict-avoiding swizzle for the 256x32 tile (passes 1-2).
__device__ __forceinline__ unsigned swz(unsigned a, unsigned c) {
  return (a << 5) | (c ^ (a & 31u));
}

// ---------------------------------------------------------------------------
// Passes 1 & 2: batched 256-pt FFTs along a strided dimension.
//   element (a, c): global index = gbase + a*gstride + c; c coalesced.
//   after FFT (a = output index k) multiply by W_twN^{(coff+c)*k}.
// ---------------------------------------------------------------------------
__global__ __launch_bounds__(TPB) void fft_pass12(const float* __restrict__ xin,
                                                  float2* __restrict__ ws,
                                                  int mode) {
  __shared__ float2 tile[LLEN * CB];  // 64 KB
  const unsigned tid = threadIdx.x;

  unsigned gbase, gstride, coff, twmask;
  double twden;
  if (mode == 0) {  // pass 1: columns m = c0..c0+31, stride M, real input
    const unsigned c0 = blockIdx.x * CB;
    gbase = c0; gstride = MDIM; coff = c0;
    twmask = FFT_N - 1u; twden = (double)FFT_N;
  } else {          // pass 2: lines (k1, n3), stride N3, complex in-place
    const unsigned k1 = blockIdx.x >> 5;           // N3/CB = 32 blocks per k1
    const unsigned n30 = (blockIdx.x & 31u) * CB;
    gbase = k1 * MDIM + n30; gstride = N3; coff = n30;
    twmask = MDIM - 1u; twden = (double)MDIM;
  }

  if (mode == 0) {
    // Real input: must synthesize imag=0, so go through VGPRs.
    for (unsigned i = tid; i < LLEN * CB; i += TPB) {
      const unsigned a = i >> 5, c = i & 31u;
      float2 v;
      v.x = xin[gbase + a * gstride + c];
      v.y = 0.0f;
      tile[swz(a, c)] = v;
    }
  } else {
    // Complex tile fill: CDNA5 async global->LDS (ASYNCcnt), 8B per lane.
    for (unsigned i = tid; i < LLEN * CB; i += TPB) {
      const unsigned a = i >> 5, c = i & 31u;
      const float2* gp = ws + ((unsigned long long)gbase + a * gstride + c);
      const unsigned loff = lds_addr32(tile + swz(a, c));
      asm volatile("global_load_async_to_lds_b64 %0, %1, off"
                   :: "v"(loff), "v"((unsigned long long)gp)
                   : "memory");
    }
    asm volatile("s_wait_asynccnt 0" ::: "memory");
  }
  __syncthreads();

  // Each wave runs 4 FFT instances (CB=32 columns / 8 waves).
  const WaveConsts wc = make_wave_consts();
  const unsigned wave = tid >> 5;
#pragma unroll 1
  for (unsigned ci = 0; ci < 4; ++ci) {
    const unsigned c = wave * 4u + ci;
    fft256_wmma(tile, wc, [c](unsigned a) { return swz(a, c); });
  }
  __syncthreads();

  // Twiddle + coalesced writeout (interleaved complex into ws).
  for (unsigned i = tid; i < LLEN * CB; i += TPB) {
    const unsigned a = i >> 5, c = i & 31u;
    float2 v = tile[swz(a, c)];
    const unsigned long long e =
        ((unsigned long long)(coff + c) * (unsigned long long)a) &
        (unsigned long long)twmask;
    const float ang = (float)(-6.283185307179586 * (double)e / twden);
    float s, cc;
    __sincosf(ang, &s, &cc);
    float2 o;
    o.x = v.x * cc - v.y * s;
    o.y = v.x * s + v.y * cc;
    ws[gbase + a * gstride + c] = o;
  }
}

// ---------------------------------------------------------------------------
// Pass 3: per line (k1,k2) a contiguous 1024-pt FFT = four stride-4 256-FFTs
// (n3 = na*4 + nb) + W1024^{nb*ka} twiddle + radix-4 over nb (k3 = ka+256*kb).
// Output: out[k1 + 256*k2 + 65536*k3] planar (re plane, im plane at +N).
// ---------------------------------------------------------------------------
__global__ __launch_bounds__(TPB) void fft_pass3(const float2* __restrict__ ws,
                                                 float* __restrict__ out) {
  __shared__ float2 tile[P3_LINES * N3];  // 64 KB
  const unsigned tid = threadIdx.x;
  const unsigned k2 = blockIdx.x & 255u;
  const unsigned a0 = (blockIdx.x >> 8) * P3_LINES;  // k1 tile base

  // Load 8 lines x 8KB, fully contiguous: async global->LDS, 16B per lane.
  for (unsigned l = 0; l < P3_LINES; ++l) {
    const unsigned q = (a0 + l) * N2 + k2;  // storage line = k1*N2 + k2
    const float2* gsrc = ws + (unsigned long long)q * N3;
    for (unsigned i = tid; i < N3 / 2u; i += TPB) {
      const unsigned loff = lds_addr32(tile + l * N3 + 2u * i);
      asm volatile("global_load_async_to_lds_b128 %0, %1, off"
                   :: "v"(loff), "v"((unsigned long long)(gsrc + 2u * i))
                   : "memory");
    }
  }
  asm volatile("s_wait_asynccnt 0" ::: "memory");
  __syncthreads();

  // 8 waves <-> 8 lines; each wave does the 4 stride-4 sub-FFTs of its line.
  const WaveConsts wc = make_wave_consts();
  const unsigned wave = tid >> 5;
#pragma unroll 1
  for (unsigned nb = 0; nb < 4; ++nb) {
    const unsigned base = wave * N3 + nb;
    fft256_wmma(tile, wc, [base](unsigned a) { return base + a * 4u; });
  }
  __syncthreads();

  // Radix-4 combine + W1024 twiddle + transposed planar writeout.
  const unsigned l  = tid & (P3_LINES - 1u);  // 0..7  -> k1 = a0 + l
  const unsigned rr = tid >> 3;               // 0..31
  const unsigned k1 = a0 + l;
  const float W1024a = -6.283185307179586f / 1024.0f;
#pragma unroll 1
  for (unsigned j = 0; j < 8; ++j) {
    const unsigned ka = rr * 8u + j;  // 0..255
    float2 tv[4];
#pragma unroll
    for (unsigned nb = 0; nb < 4; ++nb) {
      float2 b = tile[l * N3 + ka * 4u + nb];
      float s, c;
      __sincosf(W1024a * (float)(nb * ka), &s, &c);
      tv[nb].x = b.x * c - b.y * s;
      tv[nb].y = b.x * s + b.y * c;
    }
    const float2 e = make_float2(tv[0].x + tv[2].x, tv[0].y + tv[2].y);
    const float2 f = make_float2(tv[0].x - tv[2].x, tv[0].y - tv[2].y);
    const float2 g = make_float2(tv[1].x + tv[3].x, tv[1].y + tv[3].y);
    const float2 h = make_float2(tv[1].x - tv[3].x, tv[1].y - tv[3].y);
    float2 X[4];
    X[0] = make_float2(e.x + g.x, e.y + g.y);
    X[1] = make_float2(f.x + h.y, f.y - h.x);  // f + (-i)h
    X[2] = make_float2(e.x - g.x, e.y - g.y);
    X[3] = make_float2(f.x - h.y, f.y + h.x);  // f + (+i)h
#pragma unroll
    for (unsigned kb = 0; kb < 4; ++kb) {
      const unsigned k3 = ka + 256u * kb;
      const unsigned oidx = k1 + N1 * k2 + (N1 * N2) * k3;
      out[oidx] = X[kb].x;
      out[FFT_N + oidx] = X[kb].y;
    }
  }
}

// ---------------------------------------------------------------------------
extern "C" void kernel_launch(void* const* d_in, const int* in_sizes, int n_in,
                              void* d_out, int out_size, void* d_ws,
                              size_t ws_size, hipStream_t stream) {
  (void)in_sizes; (void)n_in; (void)out_size; (void)ws_size;
  const float* x = (const float*)d_in[0];
  float* out = (float*)d_out;
  float2* ws = (float2*)d_ws;  // requires ws_size >= 2^26 * 8 B = 512 MB

  fft_pass12<<<MDIM / CB, TPB, 0, stream>>>(x, ws, 0);            // 8192 blocks
  fft_pass12<<<N1 * (N3 / CB), TPB, 0, stream>>>(x, ws, 1);       // 8192 blocks
  fft_pass3 <<<(N1 * N2) / P3_LINES, TPB, 0, stream>>>(ws, out);  // 8192 blocks
}